// DarNet_64527588655134
// MI455X (gfx1250) — compile-verified
//
#include <hip/hip_runtime.h>
#include <cstddef>

// ---------------------------------------------------------------------------
// DarNet per-agent attention for MI455X (gfx1250, wave32, WMMA f32 16x16x4).
//
// Per (b,i) pair: dyn(64x16) -> q,k,v(64x16) -> scores(64x64)=q@k^T/4 ->
// softmax -> o=attn@v@Wo+bo -> relu, atomically summed over i into out.
// All matmuls are v_wmma_f32_16x16x4_f32 (K=16 == 4 chained K=4 WMMAs).
// ---------------------------------------------------------------------------

typedef float v2f __attribute__((ext_vector_type(2)));
typedef float v8f __attribute__((ext_vector_type(8)));

namespace {
constexpr int kB    = 256;   // batch
constexpr int kA    = 64;    // agents / tokens
constexpr int kMove = 4;
constexpr int kD    = 16;    // head dim
constexpr int kChunk = 80;   // ES+ALS+NACT+A per token in obs
constexpr int kObs  = 5124;  // MOVE + 5*64 + 75*64
constexpr int kOutC = kMove + kD;  // 20
}

// D = A x B + C, f32 16x16x4 (8-arg pattern: neg_a, A, neg_b, B, c_mod, C, reuse_a, reuse_b)
#define WMMA_F32X4(c, a, b) \
  (c) = __builtin_amdgcn_wmma_f32_16x16x4_f32(false, (a), false, (b), (short)0, (c), false, false)

// out[:, :, 0:4] = inputs[:, :, 0:4]; out[:, :, 4:20] = 0 (atomic accumulators)
__global__ __launch_bounds__(256) void darnet_init_out(const float* __restrict__ in,
                                                       float* __restrict__ out) {
  int idx = blockIdx.x * blockDim.x + threadIdx.x;
  constexpr int total = kB * kA * kOutC;
  if (idx >= total) return;
  int c  = idx % kOutC;
  int ba = idx / kOutC;               // b*64 + a
  out[idx] = (c < kMove) ? in[(size_t)ba * kObs + c] : 0.0f;
}

__global__ __launch_bounds__(32) void darnet_attn(
    const float* __restrict__ in,
    const float* __restrict__ Wq, const float* __restrict__ bq,
    const float* __restrict__ Wk, const float* __restrict__ bk,
    const float* __restrict__ Wv, const float* __restrict__ bv,
    const float* __restrict__ Wo, const float* __restrict__ bo,
    float* __restrict__ out)
{
  // Stride 20 (16-col mats): 20*r mod 64 distinct for r=0..15, float4-aligned rows.
  // Stride 68 (64-col attn): 68*r mod 64 = 4r, distinct for r=0..15.
  __shared__ float dynL[64][20];
  __shared__ float qL[64][20];
  __shared__ float kL[64][20];
  __shared__ float vL[64][20];
  __shared__ float wL[4][16][20];     // Wq, Wk, Wv, Wo (row-major 16x16)
  __shared__ float bL[4][16];
  __shared__ float attnL[16][68];     // one 16-row block of attn at a time
  __shared__ float oL[16][20];

  const int lane = threadIdx.x;
  const int l16  = lane & 15;
  const int half = lane >> 4;
  const int pair = blockIdx.x;        // b*64 + i
  const int b    = pair >> 6;

  const float* base = in + (size_t)pair * kObs + kMove;

  // ---- stage dyn (64 rows x 16 cols, 16 of every 80 floats) via float4 ----
  for (int e = lane; e < 256; e += 32) {
    int t  = e >> 2;
    int d4 = (e & 3) << 2;
    const float4 val = *reinterpret_cast<const float4*>(base + t * kChunk + d4);
    *reinterpret_cast<float4*>(&dynL[t][d4]) = val;
  }
  // ---- stage weights + biases ----
  const float* Ws[4] = {Wq, Wk, Wv, Wo};
  const float* Bs[4] = {bq, bk, bv, bo};
#pragma unroll
  for (int w = 0; w < 4; ++w) {
    for (int e = lane; e < 256; e += 32)
      wL[w][e >> 4][e & 15] = Ws[w][e];
    if (lane < 16) bL[w][lane] = Bs[w][lane];
  }
  __syncthreads();

  // ---- q/k/v projections: (64x16) @ (16x16) + bias, 4 row-tiles each ----
#pragma unroll
  for (int w = 0; w < 3; ++w) {
    float (*dst)[20] = (w == 0) ? qL : (w == 1) ? kL : vL;
#pragma unroll
    for (int ti = 0; ti < 4; ++ti) {
      v8f c = {};
#pragma unroll
      for (int kk = 0; kk < 4; ++kk) {
        const int kc = 4 * kk + 2 * half;
        v2f a, bf;
        a.x  = dynL[16 * ti + l16][kc];
        a.y  = dynL[16 * ti + l16][kc + 1];
        bf.x = wL[w][kc][l16];
        bf.y = wL[w][kc + 1][l16];
        WMMA_F32X4(c, a, bf);
      }
#pragma unroll
      for (int j = 0; j < 8; ++j)
        dst[16 * ti + j + 8 * half][l16] = c[j] + bL[w][l16];
    }
  }
  __syncthreads();

  // ---- per 16-row block: scores, softmax, attn@v, @Wo, relu, accumulate ----
  for (int ti = 0; ti < 4; ++ti) {
    // scores row-block: 4 tiles of q_ti @ k_si^T, scaled by 1/sqrt(16)
    v8f sc[4];
#pragma unroll
    for (int si = 0; si < 4; ++si) {
      v8f c = {};
#pragma unroll
      for (int kk = 0; kk < 4; ++kk) {
        const int kc = 4 * kk + 2 * half;
        v2f a, bf;
        a.x  = qL[16 * ti + l16][kc];
        a.y  = qL[16 * ti + l16][kc + 1];
        bf.x = kL[16 * si + l16][kc];   // B = k^T: B[d][s] = k[s][d]
        bf.y = kL[16 * si + l16][kc + 1];
        WMMA_F32X4(c, a, bf);
      }
      sc[si] = c * 0.25f;
    }

    // softmax over s (64 cols): per (j, half) row = 16 lanes x 4 si-tiles
#pragma unroll
    for (int j = 0; j < 8; ++j) {
      float m = fmaxf(fmaxf(sc[0][j], sc[1][j]), fmaxf(sc[2][j], sc[3][j]));
#pragma unroll
      for (int off = 1; off < 16; off <<= 1)
        m = fmaxf(m, __shfl_xor(m, off, 32));   // stays within 16-lane half
      float s = 0.0f;
#pragma unroll
      for (int si = 0; si < 4; ++si) {
        float e = __expf(sc[si][j] - m);
        sc[si][j] = e;
        s += e;
      }
#pragma unroll
      for (int off = 1; off < 16; off <<= 1)
        s += __shfl_xor(s, off, 32);
      const float r = 1.0f / s;
#pragma unroll
      for (int si = 0; si < 4; ++si)
        attnL[j + 8 * half][16 * si + l16] = sc[si][j] * r;
    }

    // o_ti = attn_rowblock (16x64) @ v (64x16)
    v8f co = {};
#pragma unroll
    for (int si = 0; si < 4; ++si) {
#pragma unroll
      for (int kk = 0; kk < 4; ++kk) {
        const int kc = 4 * kk + 2 * half;
        v2f a, bf;
        a.x  = attnL[l16][16 * si + kc];
        a.y  = attnL[l16][16 * si + kc + 1];
        bf.x = vL[16 * si + kc][l16];
        bf.y = vL[16 * si + kc + 1][l16];
        WMMA_F32X4(co, a, bf);
      }
    }
#pragma unroll
    for (int j = 0; j < 8; ++j)
      oL[j + 8 * half][l16] = co[j];

    // (o_ti @ Wo) + bo, relu, atomic-accumulate over agents i
    v8f cf = {};
#pragma unroll
    for (int kk = 0; kk < 4; ++kk) {
      const int kc = 4 * kk + 2 * half;
      v2f a, bf;
      a.x  = oL[l16][kc];
      a.y  = oL[l16][kc + 1];
      bf.x = wL[3][kc][l16];
      bf.y = wL[3][kc + 1][l16];
      WMMA_F32X4(cf, a, bf);
    }
#pragma unroll
    for (int j = 0; j < 8; ++j) {
      const float val = fmaxf(cf[j] + bL[3][l16], 0.0f);
      const int t = 16 * ti + j + 8 * half;
      unsafeAtomicAdd(&out[((size_t)b * kA + t) * kOutC + kMove + l16], val);
    }
  }
}

extern "C" void kernel_launch(void* const* d_in, const int* in_sizes, int n_in,
                              void* d_out, int out_size, void* d_ws, size_t ws_size,
                              hipStream_t stream) {
  (void)in_sizes; (void)n_in; (void)out_size; (void)d_ws; (void)ws_size;
  const float* in = (const float*)d_in[0];
  const float* Wq = (const float*)d_in[1];
  const float* bq = (const float*)d_in[2];
  const float* Wk = (const float*)d_in[3];
  const float* bk = (const float*)d_in[4];
  const float* Wv = (const float*)d_in[5];
  const float* bv = (const float*)d_in[6];
  const float* Wo = (const float*)d_in[7];
  const float* bo = (const float*)d_in[8];
  float* out = (float*)d_out;

  constexpr int total = kB * kA * kOutC;
  darnet_init_out<<<(total + 255) / 256, 256, 0, stream>>>(in, out);
  darnet_attn<<<kB * kA, 32, 0, stream>>>(in, Wq, bq, Wk, bk, Wv, bv, Wo, bo, out);
}